// LengthRegulator2_609885356138
// MI455X (gfx1250) — compile-verified
//
#include <hip/hip_runtime.h>
#include <stdint.h>

#define B_ 32
#define T_ 1024
#define D_ 768   // floats per frame -> 3072 bytes -> 192 float4 -> 6 iters x 32 lanes

// ---------------------------------------------------------------------------
// Kernel 1: per-row inclusive scan of durations (Hillis-Steele in LDS).
// One 1024-thread block per batch row; cost is negligible vs the 350MB copy.
// ---------------------------------------------------------------------------
__global__ __launch_bounds__(T_) void lr_scan_kernel(const int* __restrict__ dur,
                                                     int* __restrict__ cs) {
  __shared__ int s[T_];
  const int b = blockIdx.x;
  const int t = threadIdx.x;
  s[t] = dur[b * T_ + t];
  __syncthreads();
#pragma unroll
  for (int off = 1; off < T_; off <<= 1) {
    int v = (t >= off) ? s[t - off] : 0;
    __syncthreads();
    s[t] += v;
    __syncthreads();
  }
  cs[b * T_ + t] = s[t];
}

// ---------------------------------------------------------------------------
// Kernel 2: expansion. 8 waves per block, one output frame per wave.
// Binary-search the LDS cumsum, then DMA the 3072B row through LDS with the
// gfx1250 async global<->LDS path (ASYNCcnt), fully coalesced 512B/instr.
// ---------------------------------------------------------------------------
__global__ __launch_bounds__(256) void lr_expand_kernel(const float* __restrict__ x,
                                                        const int* __restrict__ cs,
                                                        float* __restrict__ out,
                                                        int max_len) {
  __shared__ int   s_cs[T_];        // 4KB: cumsum row for this batch
  __shared__ float s_stage[8][D_];  // 24KB: per-wave staging row

  const int b = blockIdx.y;
  for (int i = threadIdx.x; i < T_; i += 256) s_cs[i] = cs[b * T_ + i];
  __syncthreads();

  const int wave = threadIdx.x >> 5;
  const int lane = threadIdx.x & 31;
  const int t    = blockIdx.x * 8 + wave;   // output frame owned by this wave
  if (t >= max_len) return;

  const int total  = s_cs[T_ - 1];
  float* dstrow = out + ((size_t)b * max_len + t) * D_;

  if (t < total) {
    // first source index i with cs[i] > t  (uniform across the wave)
    int lo = 0, hi = T_;
    while (lo < hi) {
      int mid = (lo + hi) >> 1;
      if (s_cs[mid] <= t) lo = mid + 1; else hi = mid;
    }
    const float* srcrow = x + ((size_t)b * T_ + lo) * D_;

    // LDS byte offset of this wave's staging buffer (low 32b of generic ptr)
    const unsigned stage_base = (unsigned)(unsigned long long)(&s_stage[wave][0]);

    // global -> LDS, async (6 x 512B per wave)
#pragma unroll
    for (int it = 0; it < 6; ++it) {
      const int c = it * 32 + lane;                 // float4 column 0..191
      unsigned laddr = stage_base + (unsigned)c * 16u;
      uint64_t gaddr = (uint64_t)srcrow + (uint64_t)c * 16u;
      asm volatile("global_load_async_to_lds_b128 %0, %1, off"
                   :: "v"(laddr), "v"(gaddr) : "memory");
    }
    asm volatile("s_wait_asynccnt 0x0" ::: "memory");

    // LDS -> global, async; S_ENDPGM's implicit wait-idle drains the stores
#pragma unroll
    for (int it = 0; it < 6; ++it) {
      const int c = it * 32 + lane;
      unsigned laddr = stage_base + (unsigned)c * 16u;
      uint64_t gaddr = (uint64_t)dstrow + (uint64_t)c * 16u;
      asm volatile("global_store_async_from_lds_b128 %0, %1, off"
                   :: "v"(gaddr), "v"(laddr) : "memory");
    }
  } else {
    // zero padding beyond this row's expanded length
    float4 z = make_float4(0.f, 0.f, 0.f, 0.f);
    float4* d4 = (float4*)dstrow;
#pragma unroll
    for (int it = 0; it < 6; ++it) d4[it * 32 + lane] = z;
  }
}

extern "C" void kernel_launch(void* const* d_in, const int* in_sizes, int n_in,
                              void* d_out, int out_size, void* d_ws, size_t ws_size,
                              hipStream_t stream) {
  const float* x   = (const float*)d_in[0];  // [B, T, D] f32
  const int*   dur = (const int*)d_in[1];    // [B, T] i32
  float* out = (float*)d_out;                // [B, max_len, D] f32
  int*   cs  = (int*)d_ws;                   // [B, T] cumsum scratch (128KB)

  const int max_len = out_size / (B_ * D_);

  lr_scan_kernel<<<B_, T_, 0, stream>>>(dur, cs);

  dim3 grid((max_len + 7) / 8, B_);
  lr_expand_kernel<<<grid, 256, 0, stream>>>(x, cs, out, max_len);
}